// ClassBasedSMDecoder_37976100831820
// MI455X (gfx1250) — compile-verified
//
#include <hip/hip_runtime.h>

typedef __attribute__((ext_vector_type(2))) float v2f;
typedef __attribute__((ext_vector_type(8))) float v8f;

#define N_TOK 8192
#define H_DIM 1024
#define C_CLS 1024
#define K_WRD 32

#define BM 64
#define BN 128
#define BK 16
#define APAD 4
#define BPAD 4

// -------------------------------------------------------------------------
// Kernel 1: p_class[N,C] = x[N,H] @ W_cls[C,H]^T + b_cls  (fp32 WMMA GEMM)
// Double-buffered LDS, register prefetch: next stage's global loads are in
// flight while the current stage's 16 v_wmma_f32_16x16x4_f32 execute.
// -------------------------------------------------------------------------
__global__ void __launch_bounds__(256)
class_gemm_wmma(const float* __restrict__ x,
                const float* __restrict__ W,     // [C,H] row-major
                const float* __restrict__ bias,  // [C]
                float* __restrict__ out)         // [N,C]
{
    __shared__ float As[2][BK][BM + APAD];   // As[b][k][m] (K-major)
    __shared__ float Bs[2][BK][BN + BPAD];   // Bs[b][k][n] = W[n][k0+k]

    const int t    = threadIdx.x;
    const int wave = t >> 5;
    const int lane = t & 31;
    const int hi   = lane >> 4;      // 0: lanes 0-15, 1: lanes 16-31
    const int mr   = lane & 15;

    const int m_wave = (wave >> 2) * 32;   // 2 wave rows
    const int n_wave = (wave & 3) * 32;    // 4 wave cols

    const int blockM = blockIdx.y * BM;
    const int blockN = blockIdx.x * BN;

    v8f acc[2][2];
    #pragma unroll
    for (int mi = 0; mi < 2; ++mi)
        #pragma unroll
        for (int ni = 0; ni < 2; ++ni)
            acc[mi][ni] = (v8f){0.f,0.f,0.f,0.f,0.f,0.f,0.f,0.f};

    // loader mapping: one float4 of A, two float4 of B per thread per stage
    const int arow = t >> 2;              // 0..63
    const int acol = (t & 3) << 2;        // 0,4,8,12

    const float* xg  = x + (size_t)(blockM + arow) * H_DIM + acol;
    const float* wg0 = W + (size_t)(blockN + arow)      * H_DIM + acol;
    const float* wg1 = W + (size_t)(blockN + arow + 64) * H_DIM + acol;

    float4 av  = *(const float4*)(xg);
    float4 bv0 = *(const float4*)(wg0);
    float4 bv1 = *(const float4*)(wg1);

    auto store_stage = [&](int buf) {
        As[buf][acol + 0][arow] = av.x;
        As[buf][acol + 1][arow] = av.y;
        As[buf][acol + 2][arow] = av.z;
        As[buf][acol + 3][arow] = av.w;
        Bs[buf][acol + 0][arow] = bv0.x;
        Bs[buf][acol + 1][arow] = bv0.y;
        Bs[buf][acol + 2][arow] = bv0.z;
        Bs[buf][acol + 3][arow] = bv0.w;
        Bs[buf][acol + 0][arow + 64] = bv1.x;
        Bs[buf][acol + 1][arow + 64] = bv1.y;
        Bs[buf][acol + 2][arow + 64] = bv1.z;
        Bs[buf][acol + 3][arow + 64] = bv1.w;
    };

    auto prefetch = [&](int k0) {
        av  = *(const float4*)(xg  + k0);
        bv0 = *(const float4*)(wg0 + k0);
        bv1 = *(const float4*)(wg1 + k0);
    };

    auto compute_stage = [&](int buf) {
        #pragma unroll
        for (int kk = 0; kk < BK; kk += 4) {
            const int kb = kk + (hi << 1);   // lanes 16-31 hold K=+2,+3

            v2f a0, a1, b0, b1;
            a0.x = As[buf][kb    ][m_wave + mr];
            a0.y = As[buf][kb + 1][m_wave + mr];
            a1.x = As[buf][kb    ][m_wave + 16 + mr];
            a1.y = As[buf][kb + 1][m_wave + 16 + mr];
            b0.x = Bs[buf][kb    ][n_wave + mr];
            b0.y = Bs[buf][kb + 1][n_wave + mr];
            b1.x = Bs[buf][kb    ][n_wave + 16 + mr];
            b1.y = Bs[buf][kb + 1][n_wave + 16 + mr];

            acc[0][0] = __builtin_amdgcn_wmma_f32_16x16x4_f32(
                false, a0, false, b0, (short)0, acc[0][0], false, false);
            acc[0][1] = __builtin_amdgcn_wmma_f32_16x16x4_f32(
                false, a0, false, b1, (short)0, acc[0][1], false, false);
            acc[1][0] = __builtin_amdgcn_wmma_f32_16x16x4_f32(
                false, a1, false, b0, (short)0, acc[1][0], false, false);
            acc[1][1] = __builtin_amdgcn_wmma_f32_16x16x4_f32(
                false, a1, false, b1, (short)0, acc[1][1], false, false);
        }
    };

    store_stage(0);

    for (int k0 = 0; k0 < H_DIM; k0 += 2 * BK) {
        __syncthreads();
        prefetch(k0 + BK);               // always in range (loop step 2*BK)
        compute_stage(0);                // overlaps with loads above
        store_stage(1);
        __syncthreads();
        if (k0 + 2 * BK < H_DIM) prefetch(k0 + 2 * BK);
        compute_stage(1);
        if (k0 + 2 * BK < H_DIM) store_stage(0);
    }

    // C/D layout: VGPR r -> row r + 8*hi, col = lane&15
    #pragma unroll
    for (int mi = 0; mi < 2; ++mi) {
        #pragma unroll
        for (int ni = 0; ni < 2; ++ni) {
            const int gm = blockM + m_wave + mi * 16;
            const int gn = blockN + n_wave + ni * 16 + mr;
            const float bc = bias[gn];
            #pragma unroll
            for (int r = 0; r < 8; ++r) {
                out[(size_t)(gm + r + 8 * hi) * C_CLS + gn] = acc[mi][ni][r] + bc;
            }
        }
    }
}

// -------------------------------------------------------------------------
// Kernel 2: p_words[i,k] = dot(x[i,:], W_words[cls[i],k,:]) + b_words[cls[i],k]
// One block per token; x row staged in LDS via CDNA5 async global->LDS copy.
// -------------------------------------------------------------------------
__global__ void __launch_bounds__(256)
words_matvec(const float* __restrict__ x,
             const int*   __restrict__ cls,
             const float* __restrict__ W_words,  // [C,K,H]
             const float* __restrict__ b_words,  // [C,K]
             float* __restrict__ p_words)        // [N,K]
{
    __shared__ float xs[H_DIM];
    const int i = blockIdx.x;
    const int t = threadIdx.x;
    const int c = cls[i];

    // async copy: 16 bytes/lane, 256 lanes => 4KB x row, LDS write bypasses VGPRs.
    // Low 32 bits of the flat shared address are the LDS byte offset.
    {
        unsigned      lds = (unsigned)(uintptr_t)(&xs[t * 4]);
        const float*  g   = x + (size_t)i * H_DIM + t * 4;
        asm volatile("global_load_async_to_lds_b128 %0, %1, off"
                     :: "v"(lds), "v"(g) : "memory");
        asm volatile("s_wait_asynccnt 0x0" ::: "memory");
    }
    __syncthreads();

    const int wave = t >> 5;
    const int lane = t & 31;

    // each lane caches its 8 float4 chunks of x (reused for 4 outputs)
    float4 xr[8];
    #pragma unroll
    for (int j = 0; j < 8; ++j)
        xr[j] = ((const float4*)xs)[lane + 32 * j];

    const float* Wb = W_words + (size_t)c * K_WRD * H_DIM;

    #pragma unroll
    for (int q = 0; q < 4; ++q) {
        const int k = wave * 4 + q;
        const float4* Wr = (const float4*)(Wb + (size_t)k * H_DIM);
        float s = 0.f;
        #pragma unroll
        for (int j = 0; j < 8; ++j) {
            const float4 w = Wr[lane + 32 * j];
            s += xr[j].x * w.x + xr[j].y * w.y + xr[j].z * w.z + xr[j].w * w.w;
        }
        // wave32 reduction
        #pragma unroll
        for (int off = 16; off > 0; off >>= 1)
            s += __shfl_xor(s, off, 32);
        if (lane == 0)
            p_words[(size_t)i * K_WRD + k] = s + b_words[(size_t)c * K_WRD + k];
    }
}

// -------------------------------------------------------------------------
extern "C" void kernel_launch(void* const* d_in, const int* in_sizes, int n_in,
                              void* d_out, int out_size, void* d_ws, size_t ws_size,
                              hipStream_t stream)
{
    const float* x       = (const float*)d_in[0];
    const int*   cls     = (const int*)  d_in[1];
    const float* W_cls   = (const float*)d_in[2];
    const float* b_cls   = (const float*)d_in[3];
    const float* W_words = (const float*)d_in[4];
    const float* b_words = (const float*)d_in[5];

    float* p_class = (float*)d_out;                         // [N,C]
    float* p_words = p_class + (size_t)N_TOK * C_CLS;       // [N,K]

    dim3 gridA(C_CLS / BN, N_TOK / BM);   // 8 x 128 blocks
    class_gemm_wmma<<<gridA, 256, 0, stream>>>(x, W_cls, b_cls, p_class);

    words_matvec<<<N_TOK, 256, 0, stream>>>(x, cls, W_words, b_words, p_words);
}